// Affinity_Propagate_Spatial_39565238731218
// MI455X (gfx1250) — compile-verified
//
#include <hip/hip_runtime.h>
#include <stdint.h>
#include <stddef.h>

#define AS1 __attribute__((address_space(1)))
#define AS3 __attribute__((address_space(3)))

// ---------------- problem constants (from the reference) ----------------
constexpr int BATCH = 16;
constexpr int CHG   = 8;      // guidance channels == 8 neighbor gates
constexpr int H     = 512;
constexpr int W     = 512;
constexpr int HW    = H * W;

// ---------------- tiling ----------------
constexpr int TILE  = 32;             // output tile per block
constexpr int HALO  = 4;              // 4 propagation iterations => halo 4
constexpr int TS    = TILE + 2*HALO;  // 40 staged span
constexpr int NPIX  = TS * TS;        // 1600 staged pixels
constexpr int NT    = 320;            // 10 wave32s; 1600/320 = 5 pixels/thread
constexpr int PPT   = NPIX / NT;      // 5

// Neighbor offsets: shifted[i,j] = x[i+di, j+dj]
constexpr int DI[8] = { 1, 1, 1, 0, 0,-1,-1,-1};
constexpr int DJ[8] = { 1, 0,-1, 1,-1, 1, 0,-1};

// -------- gfx1250 async global->LDS copy (ASYNCcnt path) --------
__device__ __forceinline__ void async_f32_to_lds(const float* gptr, float* lptr) {
#if __has_builtin(__builtin_amdgcn_global_load_async_to_lds_b32)
  // Probe-verified prototype: (global int* src, shared int* dst, int off, int cpol)
  __builtin_amdgcn_global_load_async_to_lds_b32(
      (AS1 int*)(uintptr_t)gptr,
      (AS3 int*)(AS3 void*)lptr,
      /*offset=*/0, /*cpol=*/0);
#else
  asm volatile("global_load_async_to_lds_b32 %0, %1, off"
               :
               : "v"((uint32_t)(uintptr_t)(AS3 void*)lptr),
                 "v"((uint64_t)(uintptr_t)gptr)
               : "memory");
#endif
}

__device__ __forceinline__ void wait_asynccnt0() {
#if __has_builtin(__builtin_amdgcn_s_wait_asynccnt)
  __builtin_amdgcn_s_wait_asynccnt(0);
#else
  asm volatile("s_wait_asynccnt 0" ::: "memory");
#endif
}

extern "C" __global__ void __launch_bounds__(NT)
ap_spatial_kernel(const float* __restrict__ guidance,
                  const float* __restrict__ blur,
                  float* __restrict__ out)
{
  // Only the propagating state lives in LDS (neighbor access needs sharing);
  // gates / gate_sum / raw are register-resident (read only at owning pixel).
  __shared__ float s_raw[NPIX];
  __shared__ float s_rA [NPIX];
  __shared__ float s_rB [NPIX];

  const int tid = threadIdx.x;
  const int tx0 = blockIdx.x * TILE;
  const int ty0 = blockIdx.y * TILE;
  const int b   = blockIdx.z;

  const float* gbase = guidance + (size_t)b * CHG * HW;
  const float* dbase = blur     + (size_t)b * HW;
  float*       obase = out      + (size_t)b * HW;

  // Per-owned-pixel register state
  int   lyr[PPT], lxr[PPT];
  bool  inImg[PPT];
  float gate[PPT][CHG], gsum[PPT], raw[PPT];

  // Speculative prefetch of this thread's first guidance cacheline
  // (lowers to global_prefetch_b8 on gfx1250).
  {
    int ly = tid / TS, lx = tid - ly * TS;
    int gy = ty0 - HALO + ly + DI[0], gx = tx0 - HALO + lx + DJ[0];
    if ((unsigned)gy < (unsigned)H && (unsigned)gx < (unsigned)W)
      __builtin_prefetch(gbase + (size_t)gy * W + gx, 0, 1);
  }

  // 1) zero-fill staging tile (halo may fall outside the image)
#pragma unroll
  for (int j = 0; j < PPT; ++j) s_raw[tid + j * NT] = 0.0f;
  __syncthreads();  // ds stores retired before async writes can land

  // 2) async-stage blur_depth tile; out-of-image lanes skip -> stay zero
#pragma unroll
  for (int j = 0; j < PPT; ++j) {
    int i  = tid + j * NT;
    int ly = i / TS, lx = i - ly * TS;
    lyr[j] = ly; lxr[j] = lx;
    int gy = ty0 - HALO + ly, gx = tx0 - HALO + lx;
    bool ok = (unsigned)gy < (unsigned)H && (unsigned)gx < (unsigned)W;
    inImg[j] = ok;
    if (ok) async_f32_to_lds(dbase + (size_t)gy * W + gx, &s_raw[i]);
  }

  // 3) normalized gates straight into registers while the async copy is in
  //    flight. Each guidance element feeds exactly one (pixel,k) pair, so
  //    direct coalesced global reads are optimal (L2 absorbs halo overlap).
#pragma unroll
  for (int j = 0; j < PPT; ++j) {
    int gy = ty0 - HALO + lyr[j], gx = tx0 - HALO + lxr[j];
    float aw = 0.0f;
#pragma unroll
    for (int k = 0; k < CHG; ++k) {
      int sy = gy + DI[k], sx = gx + DJ[k];
      float v = 0.0f;
      if ((unsigned)sy < (unsigned)H && (unsigned)sx < (unsigned)W)
        v = gbase[(size_t)k * HW + (size_t)sy * W + sx];
      gate[j][k] = v;
      aw += __builtin_fabsf(v);
    }
    float inv = (aw > 0.0f) ? (1.0f / aw) : 0.0f;  // guard 0/0 in dead halo
    float gs = 0.0f;
#pragma unroll
    for (int k = 0; k < CHG; ++k) {
      gate[j][k] *= inv;
      gs += gate[j][k];
    }
    gsum[j] = gs;
  }

  wait_asynccnt0();
  __syncthreads();

  // Snapshot own raw value to a register (used by every iteration).
#pragma unroll
  for (int j = 0; j < PPT; ++j) raw[j] = s_raw[tid + j * NT];

  // 4) propagation steps 1..3, LDS-resident result, register gates.
  //    Valid region shrinks by 1/step; out-of-image pixels forced to 0
  //    (matches the reference's zero-padded shifts).
#define PROP_STEP(PREV, NEXT, T)                                              \
  _Pragma("unroll")                                                           \
  for (int j = 0; j < PPT; ++j) {                                             \
    if (lyr[j] >= (T) && lyr[j] < TS - (T) &&                                 \
        lxr[j] >= (T) && lxr[j] < TS - (T)) {                                 \
      int i = tid + j * NT;                                                   \
      float acc = 0.0f;                                                       \
      if (inImg[j]) {                                                         \
        _Pragma("unroll")                                                     \
        for (int k = 0; k < CHG; ++k)                                         \
          acc = fmaf(gate[j][k], (PREV)[i + DI[k] * TS + DJ[k]], acc);        \
        acc = fmaf(1.0f - gsum[j], raw[j], acc);                              \
      }                                                                       \
      (NEXT)[i] = acc;                                                        \
    }                                                                         \
  }                                                                           \
  __syncthreads();

  PROP_STEP(s_raw, s_rA, 1)
  PROP_STEP(s_rA,  s_rB, 2)
  PROP_STEP(s_rB,  s_rA, 3)
#undef PROP_STEP

  // 5) step 4 computes exactly the 32x32 output region -> fuse with the
  //    coalesced global store (512 % 32 == 0, so always in-image there).
#pragma unroll
  for (int j = 0; j < PPT; ++j) {
    if (lyr[j] >= HALO && lyr[j] < TS - HALO &&
        lxr[j] >= HALO && lxr[j] < TS - HALO) {
      int i  = tid + j * NT;
      int gy = ty0 - HALO + lyr[j], gx = tx0 - HALO + lxr[j];
      float acc = 0.0f;
#pragma unroll
      for (int k = 0; k < CHG; ++k)
        acc = fmaf(gate[j][k], s_rA[i + DI[k] * TS + DJ[k]], acc);
      acc = fmaf(1.0f - gsum[j], raw[j], acc);
      obase[(size_t)gy * W + gx] = acc;
    }
  }
}

extern "C" void kernel_launch(void* const* d_in, const int* in_sizes, int n_in,
                              void* d_out, int out_size, void* d_ws, size_t ws_size,
                              hipStream_t stream) {
  (void)in_sizes; (void)n_in; (void)out_size; (void)d_ws; (void)ws_size;
  const float* guidance = (const float*)d_in[0];  // [16,8,512,512] f32
  const float* blur     = (const float*)d_in[1];  // [16,1,512,512] f32
  float* out = (float*)d_out;                     // [16,1,512,512] f32

  dim3 grid(W / TILE, H / TILE, BATCH);           // (16,16,16)
  ap_spatial_kernel<<<grid, NT, 0, stream>>>(guidance, blur, out);
}